// OCDMModulator_88991722373212
// MI455X (gfx1250) — compile-verified
//
#include <hip/hip_runtime.h>

typedef __attribute__((ext_vector_type(2))) float v2f;
typedef __attribute__((ext_vector_type(8))) float v8f;

// OCDM modulate: out(row, t) = CP-extended IFFT(chirp .* sym(row, :)) * sqrt(64)
// Folded into complex GEMM with fixed 64x64 matrix M[k][n] = (1/8) exp(i*pi*(k^2+2kn)/64).
// Four real matmuls done with V_WMMA_F32_16X16X4_F32 (full fp32 precision; kernel is
// HBM-bound at ~144 MiB / 23.3 TB/s so f32 WMMA throughput is not the limiter).
__global__ __launch_bounds__(256) void ocdm_wmma_kernel(const float* __restrict__ sr,
                                                        const float* __restrict__ si,
                                                        float* __restrict__ out,
                                                        int num_rows) {
  // B operand (4x16 f32, K-split layout) staged in LDS for all 4 N-tiles, 16 K-steps.
  // Layout: [(ntile*16 + kk)*32 + lane] -> v2f {k = 4kk+2*(lane/16), +1}, n = ntile*16 + lane%16
  __shared__ float Br [4 * 16 * 32 * 2];  // Re(M)        16 KB
  __shared__ float Bi [4 * 16 * 32 * 2];  // Im(M)        16 KB
  __shared__ float Bin[4 * 16 * 32 * 2];  // -Im(M)       16 KB

  const int tid = threadIdx.x;
  const float PI_OVER_64 = 3.14159265358979323846f / 64.0f;

  // --- Build M in WMMA B-layout (4096 complex entries, 16 per thread) ---
  #pragma unroll 4
  for (int idx = tid * 16; idx < tid * 16 + 16; ++idx) {
    int k = idx >> 6;         // 0..63
    int n = idx & 63;         // 0..63
    int ph = (k * k + 2 * k * n) & 127;   // phase mod 2*pi (exact integer reduction)
    float s, c;
    __sincosf(PI_OVER_64 * (float)ph, &s, &c);
    int kk    = k >> 2;
    int e     = k & 1;
    int lane  = (n & 15) + ((k & 2) << 3);   // half-wave split on k%4 >= 2
    int ntile = n >> 4;
    int off = (((ntile * 16 + kk) * 32) + lane) * 2 + e;
    Br [off] =  0.125f * c;
    Bi [off] =  0.125f * s;
    Bin[off] = -0.125f * s;
  }
  __syncthreads();

  const int wave = tid >> 5;
  const int lane = tid & 31;
  const int rowtile = blockIdx.x * 8 + wave;
  if (rowtile * 16 >= num_rows) return;        // wave-uniform: EXEC stays all-1s for WMMA

  const int half = lane >> 4;                  // 0: lanes 0-15, 1: lanes 16-31
  const int rA = rowtile * 16 + (lane & 15);   // A-matrix row held by this lane

  // --- Load A (16x4 f32 layout): lane holds {k, k+1} with k = 4*kk + 2*half ---
  v2f Ar[16], Ai[16];
  const float* pr = sr + (size_t)rA * 64 + 2 * half;
  const float* pi = si + (size_t)rA * 64 + 2 * half;
  #pragma unroll
  for (int kk = 0; kk < 16; ++kk) {
    Ar[kk] = *(const v2f*)(pr + kk * 4);
    Ai[kk] = *(const v2f*)(pi + kk * 4);
  }

  const int rD = rowtile * 16 + 8 * half;      // D row = rD + v (v = VGPR index 0..7)

  #pragma unroll
  for (int ntile = 0; ntile < 4; ++ntile) {
    v8f cre = {};
    v8f cim = {};
    #pragma unroll
    for (int kk = 0; kk < 16; ++kk) {
      const int bo = (ntile * 16 + kk) * 32 + lane;
      v2f br  = ((const v2f*)Br )[bo];
      v2f bi  = ((const v2f*)Bi )[bo];
      v2f bin = ((const v2f*)Bin)[bo];
      // Re(D) = Ar*Br - Ai*Bi ;  Im(D) = Ar*Bi + Ai*Br
      cre = __builtin_amdgcn_wmma_f32_16x16x4_f32(false, Ar[kk], false, br,  (short)0, cre, false, false);
      cre = __builtin_amdgcn_wmma_f32_16x16x4_f32(false, Ai[kk], false, bin, (short)0, cre, false, false);
      cim = __builtin_amdgcn_wmma_f32_16x16x4_f32(false, Ar[kk], false, bi,  (short)0, cim, false, false);
      cim = __builtin_amdgcn_wmma_f32_16x16x4_f32(false, Ai[kk], false, br,  (short)0, cim, false, false);
    }

    // --- Store: output row = 80 samples * 2 floats; t = n + 16 always, plus CP t = n - 48 ---
    const int n = ntile * 16 + (lane & 15);
    #pragma unroll
    for (int v = 0; v < 8; ++v) {
      size_t base = (size_t)(rD + v) * 160;
      v2f val = { cre[v], cim[v] };
      *(v2f*)(out + base + (size_t)(n + 16) * 2) = val;
      if (ntile == 3) {
        *(v2f*)(out + base + (size_t)(n - 48) * 2) = val;   // cyclic prefix copy
      }
    }
  }
}

extern "C" void kernel_launch(void* const* d_in, const int* in_sizes, int n_in,
                              void* d_out, int out_size, void* d_ws, size_t ws_size,
                              hipStream_t stream) {
  const float* sr = (const float*)d_in[0];   // symbols_real (256,512,64) f32
  const float* si = (const float*)d_in[1];   // symbols_imag (256,512,64) f32
  float* out = (float*)d_out;                // (256, 512*80, 2) f32

  const int num_rows = in_sizes[0] / 64;           // 131072 symbol rows
  const int rowtiles = (num_rows + 15) / 16;       // 8192
  const int blocks   = (rowtiles + 7) / 8;         // 8 waves (row-tiles) per block

  ocdm_wmma_kernel<<<blocks, 256, 0, stream>>>(sr, si, out, num_rows);
}